// TensorProduct_6863357739081
// MI455X (gfx1250) — compile-verified
//
#include <hip/hip_runtime.h>

typedef __attribute__((ext_vector_type(2))) float v2f;
typedef __attribute__((ext_vector_type(8))) float v8f;

#define TILE_M  16
#define X1S     1160   // padded LDS row stride for x1 (1152 + 8)
#define GSTR    115    // per-row G coefficient block (total across 11 instructions)
#define ZSTR    132    // padded LDS row stride for z (128 + 4) -> conflict-free b64 frag loads
#define THREADS 256
#define NROW    1152   // DIM1 == DIM3
#define NCOL2   9      // DIM2
#define MROWS   50000

struct TPParams {
  const float* x1;
  const float* x2;
  float*       out;
  const float* w[11];    // 11 weight matrices, each 128x128 (u-major)
  const float* cgall;    // precomputed 115x9 coefficient matrix (in d_ws)
};

extern __shared__ float smem[];

// ---------------------------------------------------------------------------
// Setup kernel: build CGALL[115][9] so that G[m,p] = sum_j CGALL[p][j]*x2[m][j].
// Encodes all 11 instructions' CG contractions + path normalizations.
// ---------------------------------------------------------------------------
__global__ void build_cgall(const float* __restrict__ cg5, const float* __restrict__ cg6,
                            const float* __restrict__ cg8, const float* __restrict__ cg10,
                            float* __restrict__ cgall) {
  const int p = threadIdx.x;
  if (p >= GSTR) return;
  const float rs3 = 0.5773502691896258f;   // 1/sqrt(3)
  const float rs5 = 0.4472135954999579f;   // 1/sqrt(5)
  float row[9];
#pragma unroll
  for (int j = 0; j < 9; ++j) row[j] = 0.f;

  if (p == 0) {                               // idx0 (0,0,0)
    row[0] = 1.f;
  } else if (p < 4) {                         // idx1 (0,1,1): G[1+k] = x2[1+k]/sqrt3
    row[1 + (p - 1)] = rs3;
  } else if (p < 9) {                         // idx2 (0,2,2)
    row[4 + (p - 4)] = rs5;
  } else if (p < 18) {                        // idx3 (1,0,1): diag * x2[0]/sqrt3
    const int q = p - 9, k = q / 3, i = q % 3;
    if (k == i) row[0] = rs3;
  } else if (p < 21) {                        // idx4 (1,1,0): dot /sqrt3
    row[1 + (p - 18)] = rs3;
  } else if (p < 36) {                        // idx5 (1,1,2): CG (5,3,3)
    const int q = p - 21;                     // q = k*3+i
    for (int jj = 0; jj < 3; ++jj) row[1 + jj] = cg5[q * 3 + jj];
  } else if (p < 45) {                        // idx6 (1,2,1): CG (3,3,5)
    const int q = p - 36;
    for (int jj = 0; jj < 5; ++jj) row[4 + jj] = cg6[q * 5 + jj];
  } else if (p < 70) {                        // idx7 (2,0,2): diag * x2[0]/sqrt5
    const int q = p - 45, k = q / 5, i = q % 5;
    if (k == i) row[0] = rs5;
  } else if (p < 85) {                        // idx8 (2,1,1): CG (3,5,3)
    const int q = p - 70;
    for (int jj = 0; jj < 3; ++jj) row[1 + jj] = cg8[q * 3 + jj];
  } else if (p < 90) {                        // idx9 (2,2,0): dot /sqrt5
    row[4 + (p - 85)] = rs5;
  } else {                                    // idx10 (2,2,2): CG (5,5,5)
    const int q = p - 90;
    for (int jj = 0; jj < 5; ++jj) row[4 + jj] = cg10[q * 5 + jj];
  }
#pragma unroll
  for (int j = 0; j < 9; ++j) cgall[p * 9 + j] = row[j];
}

// ---------------------------------------------------------------------------
// z[m,u,k] = sum_i x1[m, xoff + u*D1 + i] * G[m, goff + k*D1 + i]
// Fused over k: x1 fragment loaded once per (m,u), produces all d3 outputs.
// D1 is compile-time so the fragment stays in registers. Counted uniform loop
// (8 iterations exactly) -> no exec-mask loop overhead.
// ---------------------------------------------------------------------------
template<int D1>
__device__ __forceinline__ void compute_z(float* __restrict__ zs,
                                          const float* __restrict__ x1s,
                                          const float* __restrict__ Gs,
                                          int tid, int d3, int xoff, int goff) {
#pragma unroll 2
  for (int it = 0; it < (TILE_M * 128) / THREADS; ++it) {
    const int e  = tid + it * THREADS;
    const int mm = e >> 7;
    const int u  = e & 127;
    const float* xr = x1s + mm * X1S + xoff + u * D1;
    float xv[D1];
#pragma unroll
    for (int i = 0; i < D1; ++i) xv[i] = xr[i];
    const float* g  = Gs + mm * GSTR + goff;
    float*       zo = zs + mm * ZSTR + u;
    for (int k = 0; k < d3; ++k) {
      float s = 0.f;
#pragma unroll
      for (int i = 0; i < D1; ++i) s += xv[i] * g[k * D1 + i];
      zo[k * (TILE_M * ZSTR)] = s;
    }
  }
}

__device__ __forceinline__ void compute_z_dispatch(float* zs, const float* x1s,
                                                   const float* Gs, int tid, int d3,
                                                   int d1, int xoff, int goff) {
  switch (d1) {   // d1 is a compile-time constant at every call site -> folds
    case 1: compute_z<1>(zs, x1s, Gs, tid, d3, xoff, goff); break;
    case 3: compute_z<3>(zs, x1s, Gs, tid, d3, xoff, goff); break;
    default: compute_z<5>(zs, x1s, Gs, tid, d3, xoff, goff); break;
  }
}

// ---------------------------------------------------------------------------
// acc[k](16x16) += z_k(16x128) * W(128 x 16-slice), 32 K-steps of wmma f32 16x16x4
// ---------------------------------------------------------------------------
template<int D3>
__device__ __forceinline__ void gemm_instr(v8f (&acc)[D3],
                                           const float* __restrict__ wgt,
                                           const float* __restrict__ zs,
                                           int lane, int nt) {
  const int hh = lane >> 4;   // half-wave: K pair {0,1} vs {2,3}
  const int n  = lane & 15;   // A row (m) and B column (w) within tile
  const float* wcol = wgt + nt * 16 + n;   // W stored as W[u*128 + w]
#pragma unroll 4
  for (int kk = 0; kk < 32; ++kk) {
    const int u = kk * 4 + 2 * hh;
    v2f b;                       // B 4x16 fragment: K rows of W
    b.x = wcol[(u + 0) * 128];
    b.y = wcol[(u + 1) * 128];
#pragma unroll
    for (int k = 0; k < D3; ++k) {
      const v2f a = *(const v2f*)(zs + (k * TILE_M + n) * ZSTR + u);  // ds_load_b64
      acc[k] = __builtin_amdgcn_wmma_f32_16x16x4_f32(
          /*neg_a=*/false, a, /*neg_b=*/false, b,
          /*c_mod=*/(short)0, acc[k], /*reuse_a=*/false, /*reuse_b=*/false);
    }
  }
}

template<int D3, int NI>
__device__ __forceinline__ void run_group(const float* __restrict__ x1s,
                                          const float* __restrict__ Gs,
                                          float* __restrict__ zs,
                                          const int (&xoffs)[NI], const int (&d1s)[NI],
                                          const int (&goffs)[NI],
                                          const float* const (&wgts)[NI],
                                          float* __restrict__ out, long m0, int off3c,
                                          int tid) {
  v8f acc[D3];
#pragma unroll
  for (int k = 0; k < D3; ++k)
#pragma unroll
    for (int r = 0; r < 8; ++r) acc[k][r] = 0.f;

  const int lane = tid & 31;
  const int nt   = tid >> 5;   // 8 waves -> 8 slices of 16 output channels

#pragma unroll
  for (int ii = 0; ii < NI; ++ii) {
    compute_z_dispatch(zs, x1s, Gs, tid, D3, d1s[ii], xoffs[ii], goffs[ii]);
    __syncthreads();
    gemm_instr<D3>(acc, wgts[ii], zs, lane, nt);
    __syncthreads();
  }

  // store D tiles: out[m, off3c + w*D3 + k]
  const int n  = lane & 15;
  const int hh = lane >> 4;
#pragma unroll
  for (int r = 0; r < 8; ++r) {
    const int mloc = r + 8 * hh;
    float* orow = out + (m0 + mloc) * NROW + off3c + (nt * 16 + n) * D3;
#pragma unroll
    for (int k = 0; k < D3; ++k) orow[k] = acc[k][r];
  }
}

__global__ __launch_bounds__(THREADS)
void tp_wmma_kernel(TPParams p) {
  float* x1s = smem;                        // 16 * 1160
  float* Gs  = smem + TILE_M * X1S;         // 16 * 115
  float* zs  = Gs + TILE_M * GSTR;          // 5 * 16 * 132

  const int  tid = threadIdx.x;
  const long m0  = (long)blockIdx.x * TILE_M;

  // ---- prefetch next tile's x1 (global_prefetch, no counter cost) ----
  if (blockIdx.x + 1 < gridDim.x) {
    const char* nx = (const char*)(p.x1 + (m0 + TILE_M) * NROW);
    for (int t = tid; t < (TILE_M * NROW * 4) / 256; t += THREADS)
      __builtin_prefetch(nx + t * 256, 0, 1);
  }

  // ---- stage x1 tile into LDS (float4), counted uniform loop (18 iters) ----
  const float4* __restrict__ x1g = (const float4*)(p.x1 + m0 * NROW);
  for (int it = 0; it < (TILE_M * (NROW / 4)) / THREADS; ++it) {
    const int e = tid + it * THREADS;
    const int m = e / (NROW / 4);
    const int c = e % (NROW / 4);
    float4 v = x1g[m * (NROW / 4) + c];
    *(float4*)(x1s + m * X1S + c * 4) = v;
  }

  // ---- G phase as dense 115x9 matvec, all 256 threads ----
  {
    const int m = tid >> 4;     // row 0..15
    const int s = tid & 15;     // 16 threads per row, 8 entries each
    float xv[9];
#pragma unroll
    for (int j = 0; j < 9; ++j) xv[j] = p.x2[(m0 + m) * NCOL2 + j];
#pragma unroll
    for (int pb = 0; pb < 8; ++pb) {
      const int pp = s * 8 + pb;
      if (pp < GSTR) {
        const float* cr = p.cgall + pp * 9;
        float a = 0.f;
#pragma unroll
        for (int j = 0; j < 9; ++j) a += cr[j] * xv[j];
        Gs[m * GSTR + pp] = a;
      }
    }
  }
  __syncthreads();

  // ---- group i3 = 0 (d3=1): instrs 0,4,9 ----
  {
    const int xo[3] = {0, 128, 512};
    const int d1[3] = {1, 3, 5};
    const int go[3] = {0, 18, 85};
    const float* const wg[3] = {p.w[0], p.w[4], p.w[9]};
    run_group<1, 3>(x1s, Gs, zs, xo, d1, go, wg, p.out, m0, 0, tid);
  }
  // ---- group i3 = 1 (d3=3): instrs 1,3,6,8 ----
  {
    const int xo[4] = {0, 128, 128, 512};
    const int d1[4] = {1, 3, 3, 5};
    const int go[4] = {1, 9, 36, 70};
    const float* const wg[4] = {p.w[1], p.w[3], p.w[6], p.w[8]};
    run_group<3, 4>(x1s, Gs, zs, xo, d1, go, wg, p.out, m0, 128, tid);
  }
  // ---- group i3 = 2 (d3=5): instrs 2,5,7,10 ----
  {
    const int xo[4] = {0, 128, 512, 512};
    const int d1[4] = {1, 3, 5, 5};
    const int go[4] = {4, 21, 45, 90};
    const float* const wg[4] = {p.w[2], p.w[5], p.w[7], p.w[10]};
    run_group<5, 4>(x1s, Gs, zs, xo, d1, go, wg, p.out, m0, 512, tid);
  }
}

extern "C" void kernel_launch(void* const* d_in, const int* in_sizes, int n_in,
                              void* d_out, int out_size, void* d_ws, size_t ws_size,
                              hipStream_t stream) {
  const float* cg5  = (const float*)d_in[13 + 5];
  const float* cg6  = (const float*)d_in[13 + 6];
  const float* cg8  = (const float*)d_in[13 + 8];
  const float* cg10 = (const float*)d_in[13 + 10];
  float* cgall = (float*)d_ws;     // 115*9 floats = 4140 bytes of scratch

  build_cgall<<<dim3(1), dim3(128), 0, stream>>>(cg5, cg6, cg8, cg10, cgall);

  TPParams p;
  p.x1 = (const float*)d_in[0];
  p.x2 = (const float*)d_in[1];
  for (int i = 0; i < 11; ++i) p.w[i] = (const float*)d_in[2 + i];
  p.cgall = cgall;
  p.out   = (float*)d_out;

  const size_t smem_bytes =
      (size_t)(TILE_M * X1S + TILE_M * GSTR + 5 * TILE_M * ZSTR) * sizeof(float);
  (void)hipFuncSetAttribute(reinterpret_cast<const void*>(tp_wmma_kernel),
                            hipFuncAttributeMaxDynamicSharedMemorySize,
                            (int)smem_bytes);
  tp_wmma_kernel<<<dim3(MROWS / TILE_M), dim3(THREADS), smem_bytes, stream>>>(p);
}